// MyRelConv_77214922048099
// MI455X (gfx1250) — compile-verified
//
#include <hip/hip_runtime.h>

#define N_NODES_C 100000
#define N_EDGES_C 1600000
#define D_C 64
#define NEG_SLOPE_C 0.2f

typedef __attribute__((ext_vector_type(2))) float v2f;
typedef __attribute__((ext_vector_type(8))) float v8f;

// ---------------------------------------------------------------------------
// float atomic max via ordered-integer trick (works across mixed signs;
// emax initialized to -inf = 0xFF800000)
// ---------------------------------------------------------------------------
__device__ __forceinline__ void atomic_max_f32(float* addr, float val) {
    if (val >= 0.0f) {
        atomicMax((int*)addr, __float_as_int(val));
    } else {
        atomicMin((unsigned int*)addr, __float_as_uint(val));
    }
}

// ---------------------------------------------------------------------------
// K1: zero / -inf initialize workspace (must run every launch: the harness
// does not re-poison between graph replays, so we own our state).
// ---------------------------------------------------------------------------
__global__ void k_init(float* __restrict__ emax, float* __restrict__ denom,
                       float* __restrict__ hneigh) {
    int i = blockIdx.x * blockDim.x + threadIdx.x;
    if (i < N_NODES_C * D_C) hneigh[i] = 0.0f;
    if (i < N_NODES_C) {
        emax[i]  = -__builtin_huge_valf();
        denom[i] = 0.0f;
    }
}

// ---------------------------------------------------------------------------
// K2: per-edge score e = leaky_relu(rel . attn_e); segment-max into emax[dst]
// ---------------------------------------------------------------------------
__global__ void k_edge_score(const float* __restrict__ rel,
                             const float* __restrict__ attn_e,
                             const int* __restrict__ dst,
                             float* __restrict__ ebuf,
                             float* __restrict__ emax) {
    int e = blockIdx.x * blockDim.x + threadIdx.x;
    if (e >= N_EDGES_C) return;
    float a0 = attn_e[0];
    float a1 = attn_e[1];
    float2 r = ((const float2*)rel)[e];          // rel is [E,2], 8B aligned
    float s = r.x * a0 + r.y * a1;
    s = (s > 0.0f) ? s : s * NEG_SLOPE_C;        // leaky_relu
    ebuf[e] = s;
    atomic_max_f32(&emax[dst[e]], s);
}

// ---------------------------------------------------------------------------
// K3: ex = exp(e - emax[dst]); segment-sum into denom[dst]
// ---------------------------------------------------------------------------
__global__ void k_edge_exp(const int* __restrict__ dst,
                           const float* __restrict__ emax,
                           float* __restrict__ ebuf,
                           float* __restrict__ denom) {
    int e = blockIdx.x * blockDim.x + threadIdx.x;
    if (e >= N_EDGES_C) return;
    int d = dst[e];
    float ex = expf(ebuf[e] - emax[d]);
    ebuf[e] = ex;
    atomicAdd(&denom[d], ex);
}

// ---------------------------------------------------------------------------
// K4: weighted gather-scatter. 16 threads per edge, one float4 (16B) per
// thread -> coalesced 256B row reads of feat[src], 4 f32 atomics into
// h_neigh[dst]. feat and h_neigh are both L2-resident (25.6 MB each).
// ---------------------------------------------------------------------------
__global__ void k_scatter(const float* __restrict__ feat,
                          const int* __restrict__ src,
                          const int* __restrict__ dst,
                          const float* __restrict__ ebuf,
                          const float* __restrict__ denom,
                          float* __restrict__ hneigh) {
    int t = blockIdx.x * blockDim.x + threadIdx.x;   // N_EDGES*16 = 25.6M < 2^31
    int e   = t >> 4;
    int sub = t & 15;
    if (e >= N_EDGES_C) return;
    int d = dst[e];
    float a = ebuf[e] / denom[d];
    const float4* fs = (const float4*)(feat + (size_t)src[e] * D_C);
    float4 v = fs[sub];
    float* hp = hneigh + (size_t)d * D_C + sub * 4;
    atomicAdd(hp + 0, a * v.x);
    atomicAdd(hp + 1, a * v.y);
    atomicAdd(hp + 2, a * v.z);
    atomicAdd(hp + 3, a * v.w);
}

// ---------------------------------------------------------------------------
// K5: out = feat @ W_self^T + h_neigh @ W_neigh^T + (b_self + b_neigh)
// One wave32 per 16x16 output tile; K=64 via 16 x V_WMMA_F32_16X16X4_F32 per
// GEMM, both GEMMs accumulated into one v8f. Full fp32 precision.
//
// Fragment layouts (ISA 7.12.2, f32 16x16x4):
//   A (16x4): lane L holds row M=L&15; lanes 0-15 carry K={k,k+1} in
//             {v0,v1}, lanes 16-31 carry K={k+2,k+3}.
//   B (4x16): symmetric: lane L holds col N=L&15; lane half selects K pair.
//   C/D (16x16, 8 VGPRs): vgpr r: lanes 0-15 -> (M=r, N=lane),
//                                  lanes 16-31 -> (M=r+8, N=lane-16).
// ---------------------------------------------------------------------------
__global__ void k_gemm_wmma(const float* __restrict__ feat,
                            const float* __restrict__ hneigh,
                            const float* __restrict__ Wself,
                            const float* __restrict__ bself,
                            const float* __restrict__ Wneigh,
                            const float* __restrict__ bneigh,
                            float* __restrict__ out) {
    const int lane    = threadIdx.x & 31;
    const int wave    = threadIdx.x >> 5;
    const int nColT   = D_C / 16;                        // 4
    const int nRowT   = N_NODES_C / 16;                  // 6250 (exact)
    int tile = blockIdx.x * (blockDim.x >> 5) + wave;
    if (tile >= nRowT * nColT) return;                   // wave-uniform guard

    const int rt = tile / nColT;
    const int ct = tile - rt * nColT;
    const int m0 = rt * 16;
    const int n0 = ct * 16;

    const int mn   = lane & 15;        // A-row (M) and B-col (N=j of W) index
    const int koff = (lane >> 4) * 2;  // lane half selects K pair: {0,1} / {2,3}

    const float* arow0 = feat   + (size_t)(m0 + mn) * D_C;
    const float* brow0 = Wself  + (size_t)(n0 + mn) * D_C;   // B[k][j] = W[j*64+k]
    const float* arow1 = hneigh + (size_t)(m0 + mn) * D_C;
    const float* brow1 = Wneigh + (size_t)(n0 + mn) * D_C;

    v8f c = {};
#pragma unroll
    for (int kk = 0; kk < D_C; kk += 4) {
        v2f a, b;
        a.x = arow0[kk + koff]; a.y = arow0[kk + koff + 1];
        b.x = brow0[kk + koff]; b.y = brow0[kk + koff + 1];
        c = __builtin_amdgcn_wmma_f32_16x16x4_f32(
                /*neg_a=*/false, a, /*neg_b=*/false, b,
                /*c_mod=*/(short)0, c, /*reuse_a=*/false, /*reuse_b=*/false);
    }
#pragma unroll
    for (int kk = 0; kk < D_C; kk += 4) {
        v2f a, b;
        a.x = arow1[kk + koff]; a.y = arow1[kk + koff + 1];
        b.x = brow1[kk + koff]; b.y = brow1[kk + koff + 1];
        c = __builtin_amdgcn_wmma_f32_16x16x4_f32(
                false, a, false, b, (short)0, c, false, false);
    }

    const int col  = n0 + mn;
    const float bias = bself[col] + bneigh[col];
    const int rbase = m0 + ((lane >> 4) ? 8 : 0);
#pragma unroll
    for (int r = 0; r < 8; ++r) {
        out[(size_t)(rbase + r) * D_C + col] = c[r] + bias;
    }
}

// ---------------------------------------------------------------------------
extern "C" void kernel_launch(void* const* d_in, const int* in_sizes, int n_in,
                              void* d_out, int out_size, void* d_ws, size_t ws_size,
                              hipStream_t stream) {
    const float* feat    = (const float*)d_in[0];
    const float* rel     = (const float*)d_in[1];
    const float* W_self  = (const float*)d_in[2];
    const float* b_self  = (const float*)d_in[3];
    const float* W_neigh = (const float*)d_in[4];
    const float* b_neigh = (const float*)d_in[5];
    const float* attn_e  = (const float*)d_in[6];
    const int*   src     = (const int*)d_in[7];
    const int*   dst     = (const int*)d_in[8];
    float* out = (float*)d_out;

    // Workspace layout (256B-aligned slices): emax[N], denom[N], ebuf[E],
    // hneigh[N*64]  => ~32.8 MB total.
    char* ws = (char*)d_ws;
    size_t off = 0;
    auto take = [&](size_t bytes) -> void* {
        void* p = ws + off;
        off += (bytes + 255) & ~(size_t)255;
        return p;
    };
    float* emax   = (float*)take((size_t)N_NODES_C * sizeof(float));
    float* denom  = (float*)take((size_t)N_NODES_C * sizeof(float));
    float* ebuf   = (float*)take((size_t)N_EDGES_C * sizeof(float));
    float* hneigh = (float*)take((size_t)N_NODES_C * D_C * sizeof(float));
    (void)ws_size; (void)in_sizes; (void)n_in; (void)out_size;

    const int B = 256;

    // K1: init (covers N*64 elements; first N threads also init emax/denom)
    k_init<<<(N_NODES_C * D_C + B - 1) / B, B, 0, stream>>>(emax, denom, hneigh);

    // K2: edge scores + segment max
    k_edge_score<<<(N_EDGES_C + B - 1) / B, B, 0, stream>>>(rel, attn_e, dst, ebuf, emax);

    // K3: exp + segment sum
    k_edge_exp<<<(N_EDGES_C + B - 1) / B, B, 0, stream>>>(dst, emax, ebuf, denom);

    // K4: weighted gather-scatter (16 lanes per edge)
    k_scatter<<<(N_EDGES_C * 16 + B - 1) / B, B, 0, stream>>>(feat, src, dst, ebuf,
                                                              denom, hneigh);

    // K5: fused dual GEMM + bias via f32 WMMA; 25000 tiles, 4 waves/block
    const int nTiles = (N_NODES_C / 16) * (D_C / 16);
    const int wavesPerBlock = 4;
    k_gemm_wmma<<<(nTiles + wavesPerBlock - 1) / wavesPerBlock, wavesPerBlock * 32,
                  0, stream>>>(feat, hneigh, W_self, b_self, W_neigh, b_neigh, out);
}